// SparseLinear_38525856645424
// MI455X (gfx1250) — compile-verified
//
#include <hip/hip_runtime.h>

// CDNA5 wave32 WMMA fragment types
typedef __attribute__((ext_vector_type(2))) float v2f;
typedef __attribute__((ext_vector_type(8))) float v8f;

constexpr int KDIM = 4096;   // in_features
constexpr int NDIM = 4096;   // out_features
constexpr int BM   = 128;    // workgroup M tile
constexpr int BN   = 128;    // workgroup N tile
constexpr int BK   = 16;     // K block
constexpr int LDT  = BK + 4; // LDS row stride in floats (20): float4-aligned, conflict-free

// out[M,N] = x[M,K] @ (W .* mask)[N,K]^T + bias[N]
__global__ __launch_bounds__(256) void sparse_linear_wmma(
    const float* __restrict__ X,
    const float* __restrict__ W,
    const unsigned char* __restrict__ Mask,
    const float* __restrict__ Bias,
    float* __restrict__ Out)
{
    // double-buffered tiles: 2 * 2 * 128*20*4 B = 40960 B LDS
    __shared__ float xs[2][BM * LDT];
    __shared__ float ws[2][BN * LDT];

    const int tid  = threadIdx.x;
    const int wave = tid >> 5;       // 0..7
    const int lane = tid & 31;
    const int wm   = wave >> 1;      // 0..3 : 32-row slab
    const int wn   = wave & 1;       // 0..1 : 64-col slab
    const int r    = lane & 15;      // fragment row within 16
    const int kk   = (lane >> 4) << 1; // lanes 16..31 hold K+2,K+3

    const int bm = blockIdx.x * BM;
    const int bn = blockIdx.y * BN;

    // ---- staging registers for one K tile (global -> regs -> LDS) ----
    float4 xr[2], wr[2];
    auto load_tiles = [&](int k0) {
        #pragma unroll
        for (int i = 0; i < 2; ++i) {
            const int id  = tid + i * 256;   // 0..511
            const int row = id >> 2;         // 0..127
            const int kc  = (id & 3) << 2;   // 0,4,8,12
            xr[i] = *(const float4*)(X + (size_t)(bm + row) * KDIM + k0 + kc);
            float4 w = *(const float4*)(W + (size_t)(bn + row) * KDIM + k0 + kc);
            // fuse sparsity mask (1 byte per element) at load time
            const unsigned int mb =
                *(const unsigned int*)(Mask + (size_t)(bn + row) * KDIM + k0 + kc);
            w.x = (mb & 0x000000FFu) ? w.x : 0.0f;
            w.y = (mb & 0x0000FF00u) ? w.y : 0.0f;
            w.z = (mb & 0x00FF0000u) ? w.z : 0.0f;
            w.w = (mb & 0xFF000000u) ? w.w : 0.0f;
            wr[i] = w;
        }
    };
    auto store_tiles = [&](int buf) {
        #pragma unroll
        for (int i = 0; i < 2; ++i) {
            const int id  = tid + i * 256;
            const int row = id >> 2;
            const int kc  = (id & 3) << 2;
            *(float4*)&xs[buf][row * LDT + kc] = xr[i];
            *(float4*)&ws[buf][row * LDT + kc] = wr[i];
        }
    };

    // ---- accumulators: 2 (M) x 4 (N) tiles of 16x16 f32 ----
    const v8f zero = {0.f, 0.f, 0.f, 0.f, 0.f, 0.f, 0.f, 0.f};
    v8f acc[2][4];
    #pragma unroll
    for (int mt = 0; mt < 2; ++mt)
        #pragma unroll
        for (int nt = 0; nt < 4; ++nt)
            acc[mt][nt] = zero;

    load_tiles(0);
    store_tiles(0);
    __syncthreads();

    constexpr int KTILES = KDIM / BK; // 256
    for (int kt = 0; kt < KTILES; ++kt) {
        const int buf = kt & 1;
        if (kt + 1 < KTILES)
            load_tiles((kt + 1) * BK);   // prefetch next tile into registers

        const float* xb = &xs[buf][0];
        const float* wb = &ws[buf][0];
        #pragma unroll
        for (int k = 0; k < BK; k += 4) {
            v2f a[2], b[4];
            #pragma unroll
            for (int mt = 0; mt < 2; ++mt)
                a[mt] = *(const v2f*)&xb[(wm * 32 + mt * 16 + r) * LDT + k + kk];
            #pragma unroll
            for (int nt = 0; nt < 4; ++nt)
                b[nt] = *(const v2f*)&wb[(wn * 64 + nt * 16 + r) * LDT + k + kk];
            #pragma unroll
            for (int mt = 0; mt < 2; ++mt)
                #pragma unroll
                for (int nt = 0; nt < 4; ++nt)
                    acc[mt][nt] = __builtin_amdgcn_wmma_f32_16x16x4_f32(
                        /*neg_a=*/false, a[mt],
                        /*neg_b=*/false, b[nt],
                        /*c_mod=*/(short)0, acc[mt][nt],
                        /*reuse_a=*/false, /*reuse_b=*/false);
        }
        __syncthreads();
        if (kt + 1 < KTILES) {
            store_tiles(buf ^ 1);
            __syncthreads();
        }
    }

    // ---- epilogue: bias add + store ----
    // D layout: VGPR i -> row (i + (lane<16?0:8)), col = tileN + (lane&15)
    #pragma unroll
    for (int nt = 0; nt < 4; ++nt) {
        const int col = bn + wn * 64 + nt * 16 + r;
        const float bv = Bias[col];
        #pragma unroll
        for (int mt = 0; mt < 2; ++mt) {
            const int row0 = bm + wm * 32 + mt * 16 + ((lane < 16) ? 0 : 8);
            float* op = Out + (size_t)row0 * NDIM + col;
            v8f c = acc[mt][nt];
            #pragma unroll
            for (int i = 0; i < 8; ++i)
                op[(size_t)i * NDIM] = c[i] + bv;
        }
    }
}

extern "C" void kernel_launch(void* const* d_in, const int* in_sizes, int n_in,
                              void* d_out, int out_size, void* d_ws, size_t ws_size,
                              hipStream_t stream) {
    const float*         x    = (const float*)d_in[0];
    const float*         w    = (const float*)d_in[1];
    const unsigned char* mask = (const unsigned char*)d_in[2]; // jax bool -> 1 byte
    const float*         bias = (const float*)d_in[3];
    float*               out  = (float*)d_out;

    const int M = in_sizes[0] / KDIM; // n_tokens = 8192
    dim3 grid(M / BM, NDIM / BN);     // (64, 32)
    hipLaunchKernelGGL(sparse_linear_wmma, grid, dim3(256), 0, stream,
                       x, w, mask, bias, out);
}